// RoiTrainingModel_18794776887348
// MI455X (gfx1250) — compile-verified
//
#include <hip/hip_runtime.h>
#include <hip/hip_bf16.h>
#include <cmath>

// MI455X (gfx1250) implementation. Workload is latency-bound (~60 MFLOP,
// ~1.3 MB traffic << 23.3 TB/s), so structure for minimal passes, wave32
// shuffles, LDS staging, L2 prefetch warming, and a WMMA-based exact f32
// final reduction (v_wmma_f32_16x16x4_f32 with all-ones A => column sums).

#define N_PROPS     65536
#define N_GTB       64
#define NUM_CLS     81
#define TOTAL_ROI   512
#define MAX_POS_ROI 128
#define POS_THR_F   0.5f
#define NEG_THR_F   0.1f

typedef __attribute__((ext_vector_type(2))) float v2f;
typedef __attribute__((ext_vector_type(8))) float v8f;

// ---------------- Kernel A: IoU max/argmax + flags + per-block counts -------
__global__ void kA_iou(const float4* __restrict__ prop,
                       const float4* __restrict__ gtb,
                       int* __restrict__ flag, int* __restrict__ gidx,
                       int* __restrict__ posCnt, int* __restrict__ negCnt)
{
    __shared__ float4 sGT[N_GTB];
    __shared__ float  sArea[N_GTB];
    __shared__ int    sPos, sNeg;
    const int tid = threadIdx.x;
    if (tid == 0) { sPos = 0; sNeg = 0; }
    if (tid < N_GTB) {
        float4 b = gtb[tid];
        sGT[tid]   = b;
        sArea[tid] = (b.z - b.x) * (b.w - b.y);
    }
    __syncthreads();

    const int p = blockIdx.x * blockDim.x + tid;   // grid sized exactly N_PROPS
    float4 a = prop[p];
    const float areaA = (a.z - a.x) * (a.w - a.y);
    float best = -1.0f; int bg = 0;
    #pragma unroll 4
    for (int g = 0; g < N_GTB; ++g) {
        float4 b = sGT[g];
        float w = fminf(a.z, b.z) - fmaxf(a.x, b.x);
        float h = fminf(a.w, b.w) - fmaxf(a.y, b.y);
        w = fmaxf(w, 0.0f); h = fmaxf(h, 0.0f);
        float inter = w * h;
        float iou = inter / (areaA + sArea[g] - inter + 1e-8f);
        if (iou > best) { best = iou; bg = g; }   // strict > keeps first argmax
    }
    int f = (best >= POS_THR_F) ? 2 : ((best >= NEG_THR_F) ? 1 : 0);
    flag[p] = f;
    gidx[p] = bg;
    if (f == 2)      atomicAdd(&sPos, 1);
    else if (f == 1) atomicAdd(&sNeg, 1);
    __syncthreads();
    if (tid == 0) { posCnt[blockIdx.x] = sPos; negCnt[blockIdx.x] = sNeg; }
}

// ---------------- Kernel B: exclusive scan of 256 block counts --------------
__global__ void kB_scan(const int* __restrict__ posCnt, const int* __restrict__ negCnt,
                        int* __restrict__ posOff, int* __restrict__ negOff,
                        int* __restrict__ totals)
{
    __shared__ int sp[256], sn[256];
    const int tid = threadIdx.x;
    const int op = posCnt[tid], on = negCnt[tid];
    sp[tid] = op; sn[tid] = on;
    __syncthreads();
    for (int off = 1; off < 256; off <<= 1) {
        int vp = (tid >= off) ? sp[tid - off] : 0;
        int vn = (tid >= off) ? sn[tid - off] : 0;
        __syncthreads();
        sp[tid] += vp; sn[tid] += vn;
        __syncthreads();
    }
    posOff[tid] = sp[tid] - op;
    negOff[tid] = sn[tid] - on;
    if (tid == 255) {
        int pt = sp[255], nt = sn[255];
        totals[0] = pt;
        totals[1] = nt;
        totals[2] = (pt < MAX_POS_ROI) ? pt : MAX_POS_ROI;  // num_pos
    }
}

// ---------------- Kernel C: stable scatter into 512 slots + L2 prefetch ----
__global__ void kC_scatter(const int* __restrict__ flag, const int* __restrict__ gidx,
                           const int* __restrict__ posOff, const int* __restrict__ negOff,
                           const int* __restrict__ totals,
                           int* __restrict__ selIdx, int* __restrict__ selGt,
                           int* __restrict__ selPos,
                           const float* __restrict__ roiScore,
                           const float* __restrict__ roiBox)
{
    __shared__ int sp[256], sn[256];
    const int tid = threadIdx.x;
    const int p   = blockIdx.x * 256 + tid;
    const int f   = flag[p];
    const int fp  = (f == 2) ? 1 : 0;
    const int fn  = (f == 1) ? 1 : 0;
    sp[tid] = fp; sn[tid] = fn;
    __syncthreads();
    for (int off = 1; off < 256; off <<= 1) {
        int vp = (tid >= off) ? sp[tid - off] : 0;
        int vn = (tid >= off) ? sn[tid - off] : 0;
        __syncthreads();
        sp[tid] += vp; sn[tid] += vn;
        __syncthreads();
    }
    const int posRank  = posOff[blockIdx.x] + sp[tid] - fp;  // #pos with idx < p
    const int negRank  = negOff[blockIdx.x] + sn[tid] - fn;  // #neg with idx < p
    const int numPos   = totals[2];
    const int negTotal = totals[1];
    const int g = gidx[p];

    // pos_order path: slots [0, numPos)
    if (fp && posRank < numPos) {
        selIdx[posRank] = p; selGt[posRank] = g; selPos[posRank] = 1;
        __builtin_prefetch(roiScore + (size_t)p * NUM_CLS, 0, 3);
        __builtin_prefetch(roiBox + ((size_t)p * NUM_CLS + g) * 4, 0, 3);
    }
    // neg_order path: negatives first (ascending), then non-negatives (ascending)
    const int s2 = fn ? (numPos + negRank)
                      : (numPos + negTotal + (p - negRank)); // non-neg fallback rank
    if (s2 < TOTAL_ROI) {
        selIdx[s2] = p; selGt[s2] = g; selPos[s2] = 0;       // s2 >= numPos => label 0
        __builtin_prefetch(roiScore + (size_t)p * NUM_CLS, 0, 3);
    }
}

// ---------------- Kernel D: per-slot CE (one wave32/slot) + smooth-L1 -------
__global__ void kD_loss(const int* __restrict__ selIdx, const int* __restrict__ selGt,
                        const int* __restrict__ selPos,
                        const float* __restrict__ roiScore,
                        const float* __restrict__ roiBox,
                        const float4* __restrict__ prop,
                        const float4* __restrict__ gtb,
                        const int* __restrict__ gtLabels,
                        float* __restrict__ ceArr, float* __restrict__ regArr)
{
    const int lane = threadIdx.x & 31;
    const int wv   = threadIdx.x >> 5;
    const int s    = blockIdx.x * 8 + wv;           // 64 blocks * 8 waves = 512 slots
    const int i     = selIdx[s];
    const int g     = selGt[s];
    const int ispos = selPos[s];
    const int label = ispos ? gtLabels[g] : 0;

    const float* row = roiScore + (size_t)i * NUM_CLS;
    const float v0 = row[lane];
    const float v1 = row[lane + 32];
    const bool  has2 = (lane + 64) < NUM_CLS;
    const float v2 = has2 ? row[lane + 64] : -INFINITY;

    float m = fmaxf(fmaxf(v0, v1), v2);
    for (int o = 16; o; o >>= 1) m = fmaxf(m, __shfl_xor(m, o, 32));

    float se = expf(v0 - m) + expf(v1 - m) + (has2 ? expf(v2 - m) : 0.0f);
    float pick = 0.0f;
    if (label == lane)            pick = v0;
    else if (label == lane + 32)  pick = v1;
    else if (label == lane + 64)  pick = v2;
    for (int o = 16; o; o >>= 1) {
        se   += __shfl_xor(se,   o, 32);
        pick += __shfl_xor(pick, o, 32);
    }

    if (lane == 0) {
        ceArr[s] = -(pick - m - logf(se));
        float r = 0.0f;
        if (ispos) {
            float4 P = prop[i]; float4 G = gtb[g];
            float pw = P.z - P.x, ph = P.w - P.y;
            float pcx = P.x + 0.5f * pw, pcy = P.y + 0.5f * ph;
            float gw = G.z - G.x, gh = G.w - G.y;
            float gcx = G.x + 0.5f * gw, gcy = G.y + 0.5f * gh;
            float t[4];
            t[0] = ((gcx - pcx) / pw) / 0.1f;   // (t - mean)/std, mean = 0
            t[1] = ((gcy - pcy) / ph) / 0.1f;
            t[2] = logf(gw / pw) / 0.2f;
            t[3] = logf(gh / ph) / 0.2f;
            const float* pr = roiBox + ((size_t)i * NUM_CLS + g) * 4;
            #pragma unroll
            for (int d = 0; d < 4; ++d) {
                float diff = pr[d] - t[d];
                float ad = fabsf(diff);
                r += (ad < 1.0f) ? 0.5f * diff * diff : (ad - 0.5f);  // sigma = 1
            }
        }
        regArr[s] = r;   // mask is_pos folded in (0 for negatives)
    }
}

// ---------------- Kernel E: exact f32 reduction via WMMA 16x16x4 ------------
// A = all-ones (16x4) => D[m][n] = sum_k B[k][n]; every row of D equals the
// column-sum vector, so sum over all 32 lanes of D-VGPR0 = 2 * total(B).
// Layout-independent: any bijective (lane,slot)->(k,n) packing gives the same
// total. 8 chained WMMAs reduce 512 f32 values at full f32 precision.
__global__ void kE_final(const float* __restrict__ ceArr,
                         const float* __restrict__ regArr,
                         float* __restrict__ out)
{
    const int lane = threadIdx.x;   // launched with exactly 32 threads, EXEC all-1s
    v2f a; a.x = 1.0f; a.y = 1.0f;
    v8f cc = {}; v8f cr = {};
    #pragma unroll
    for (int t = 0; t < 8; ++t) {
        v2f bc, br;
        bc.x = ceArr[t * 64 + lane];
        bc.y = ceArr[t * 64 + 32 + lane];
        br.x = regArr[t * 64 + lane];
        br.y = regArr[t * 64 + 32 + lane];
        cc = __builtin_amdgcn_wmma_f32_16x16x4_f32(false, a, false, bc,
                                                   (short)0, cc, false, false);
        cr = __builtin_amdgcn_wmma_f32_16x16x4_f32(false, a, false, br,
                                                   (short)0, cr, false, false);
    }
    float sc = cc[0];
    float sr = cr[0];
    for (int o = 16; o; o >>= 1) {
        sc += __shfl_xor(sc, o, 32);
        sr += __shfl_xor(sr, o, 32);
    }
    if (lane == 0) {
        out[0] = (sc * 0.5f) / (float)TOTAL_ROI;   // CLS_W = 1, mean over 512
        out[1] = (sr * 0.5f) / (float)TOTAL_ROI;   // REG_W = 1, sum / 512
    }
}

// ---------------- Launcher --------------------------------------------------
extern "C" void kernel_launch(void* const* d_in, const int* in_sizes, int n_in,
                              void* d_out, int out_size, void* d_ws, size_t ws_size,
                              hipStream_t stream)
{
    // inputs (setup_inputs order): image_shape, rpn_proposals_bboxes, roi_score,
    //                              roi_bboxes_txtytwth, gt_bboxes, gt_labels
    const float4* prop     = (const float4*)d_in[1];
    const float*  roiScore = (const float*)d_in[2];
    const float*  roiBox   = (const float*)d_in[3];
    const float4* gtb      = (const float4*)d_in[4];
    const int*    gtLabels = (const int*)d_in[5];
    float* out = (float*)d_out;

    // workspace layout (~540 KB total, all regions written before read)
    char* ws = (char*)d_ws;
    int*   flag   = (int*)(ws + 0);        // [65536]
    int*   gidx   = (int*)(ws + 262144);   // [65536]
    int*   posCnt = (int*)(ws + 524288);   // [256]
    int*   negCnt = (int*)(ws + 525312);   // [256]
    int*   posOff = (int*)(ws + 526336);   // [256]
    int*   negOff = (int*)(ws + 527360);   // [256]
    int*   totals = (int*)(ws + 528384);   // [4] posTotal, negTotal, numPos
    int*   selIdx = (int*)(ws + 528400);   // [512]
    int*   selGt  = (int*)(ws + 530448);   // [512]
    int*   selPos = (int*)(ws + 532496);   // [512]
    float* ceArr  = (float*)(ws + 534544); // [512]
    float* regArr = (float*)(ws + 536592); // [512]

    kA_iou    <<<N_PROPS / 256, 256, 0, stream>>>(prop, gtb, flag, gidx, posCnt, negCnt);
    kB_scan   <<<1, 256, 0, stream>>>(posCnt, negCnt, posOff, negOff, totals);
    kC_scatter<<<N_PROPS / 256, 256, 0, stream>>>(flag, gidx, posOff, negOff, totals,
                                                  selIdx, selGt, selPos, roiScore, roiBox);
    kD_loss   <<<TOTAL_ROI / 8, 256, 0, stream>>>(selIdx, selGt, selPos, roiScore, roiBox,
                                                  prop, gtb, gtLabels, ceArr, regArr);
    kE_final  <<<1, 32, 0, stream>>>(ceArr, regArr, out);
    (void)in_sizes; (void)n_in; (void)out_size; (void)ws_size;
}